// TemEmbedding_45672682225716
// MI455X (gfx1250) — compile-verified
//
#include <hip/hip_runtime.h>
#include <hip/hip_bf16.h>

// Problem constants (from the reference):
//   B=8, TIMES=288, N=207, D=64, FEA_IN=295, ROWS = B*TIMES = 2304
// TE flat layout: [b][t][n][c] -> ((b*288+t)*207 + n)*2 + c ; we use n==0 only,
// so row r reads TE[r*414 + 0..1].

#define ROWS      2304
#define DDIM      64
#define NREP      207
#define ROW_FLOATS (NREP * DDIM)          // 13248 floats per output row block
#define ROW_F4     (ROW_FLOATS / 4)       // 3312 float4 per row block

typedef __attribute__((ext_vector_type(2))) float v2f;
typedef __attribute__((ext_vector_type(4))) float v4f;
typedef __attribute__((ext_vector_type(8))) float v8f;

// ---------------------------------------------------------------------------
// Stage 1: h[r][d] = relu(W1[dow_r][d] + W1[7+tod_r][d] + b1[d])
// One-hot @ W1 collapses to a 2-row gather of W1. 2304*64 threads.
// ---------------------------------------------------------------------------
__global__ void embed_h_kernel(const int* __restrict__ TE,
                               const float* __restrict__ W1,
                               const float* __restrict__ b1,
                               float* __restrict__ h) {
    const int idx = blockIdx.x * blockDim.x + threadIdx.x;   // 0 .. 147455
    const int row = idx >> 6;                                // /64
    const int d   = idx & 63;
    if (row >= ROWS) return;

    int te0 = TE[row * (NREP * 2) + 0];
    int te1 = TE[row * (NREP * 2) + 1];
    int dow = te0 % 7;   if (dow < 0) dow += 7;
    int tod = te1 % 288; if (tod < 0) tod += 288;

    float v = W1[dow * DDIM + d] + W1[(7 + tod) * DDIM + d] + b1[d];
    h[row * DDIM + d] = v > 0.0f ? v : 0.0f;
}

// ---------------------------------------------------------------------------
// Stage 2: out_r = h @ W2 + b2  (M=2304, N=64, K=64), fp32 WMMA.
// One wave computes one 16x16 tile via 16 chained V_WMMA_F32_16X16X4_F32.
// Fragment layouts per CDNA5 ISA 7.12.2:
//   A 16x4  : M = lane%16 ; VGPR v holds K = 2*(lane/16) + v
//   B 4x16  : N = lane%16 ; VGPR v holds K = 2*(lane/16) + v
//   C/D 16x16: N = lane%16 ; VGPR r holds M = r + 8*(lane/16)
// blockDim = 128 (4 waves, wave w -> n-tile w). 144 blocks cover M.
// ---------------------------------------------------------------------------
__global__ void gemm_wmma_kernel(const float* __restrict__ h,
                                 const float* __restrict__ W2,
                                 const float* __restrict__ b2,
                                 float* __restrict__ outr) {
    const int lane = threadIdx.x & 31;
    const int wave = threadIdx.x >> 5;        // 0..3 -> n tile
    const int m0   = blockIdx.x * 16;
    const int n0   = wave * 16;
    const int nl   = lane & 15;               // N (and A's M) within tile
    const int kh   = (lane >> 4) * 2;         // K offset from lane half: 0 or 2

    const float* __restrict__ arow = h + (m0 + nl) * DDIM;   // A: row m0+lane%16

    v8f c = {};
#pragma unroll
    for (int k0 = 0; k0 < DDIM; k0 += 4) {
        v2f a, b;
        a[0] = arow[k0 + kh + 0];
        a[1] = arow[k0 + kh + 1];
        b[0] = W2[(k0 + kh + 0) * DDIM + n0 + nl];
        b[1] = W2[(k0 + kh + 1) * DDIM + n0 + nl];
        // (neg_a, A, neg_b, B, c_mod, C, reuse_a, reuse_b)
        c = __builtin_amdgcn_wmma_f32_16x16x4_f32(
                false, a, false, b, (short)0, c, false, false);
    }

    const float bias  = b2[n0 + nl];
    const int   mbase = m0 + (lane >> 4) * 8;
#pragma unroll
    for (int r = 0; r < 8; ++r) {
        outr[(mbase + r) * DDIM + n0 + nl] = c[r] + bias;
    }
}

// ---------------------------------------------------------------------------
// Stage 3: broadcast each 64-float row 207x -> d_out (~116 MiB, the roofline).
// One block per row. 256 % 16 == 0, so each thread's source float4 index
// (i & 15) is loop-invariant: load once, then stream B128 non-temporal stores.
// Row block = 13248 floats = 52992 bytes (16B aligned).
// ---------------------------------------------------------------------------
__global__ void broadcast_kernel(const float* __restrict__ outr,
                                 float* __restrict__ out) {
    const int r   = blockIdx.x;     // 0..2303
    const int tid = threadIdx.x;    // 0..255

    const v4f v = ((const v4f*)(outr + r * DDIM))[tid & 15];
    v4f* __restrict__ dst = (v4f*)(out + (size_t)r * ROW_FLOATS);

    for (int i = tid; i < ROW_F4; i += 256) {
        __builtin_nontemporal_store(v, dst + i);
    }
}

// ---------------------------------------------------------------------------
extern "C" void kernel_launch(void* const* d_in, const int* in_sizes, int n_in,
                              void* d_out, int out_size, void* d_ws, size_t ws_size,
                              hipStream_t stream) {
    (void)in_sizes; (void)n_in; (void)out_size; (void)ws_size;

    const int*   TE = (const int*)  d_in[0];
    const float* W1 = (const float*)d_in[1];
    const float* b1 = (const float*)d_in[2];
    const float* W2 = (const float*)d_in[3];
    const float* b2 = (const float*)d_in[4];
    // d_in[5] is T==288 (compile-time constant here)

    float* out  = (float*)d_out;
    float* h    = (float*)d_ws;              // ROWS*DDIM floats
    float* outr = h + ROWS * DDIM;           // ROWS*DDIM floats

    // Stage 1: gather + relu -> h
    embed_h_kernel<<<(ROWS * DDIM) / 256, 256, 0, stream>>>(TE, W1, b1, h);

    // Stage 2: fp32 WMMA GEMM -> outr
    gemm_wmma_kernel<<<ROWS / 16, 128, 0, stream>>>(h, W2, b2, outr);

    // Stage 3: broadcast rows 207x -> d_out
    broadcast_kernel<<<ROWS, 256, 0, stream>>>(outr, out);
}